// STFT_62036507623563
// MI455X (gfx1250) — compile-verified
//
#include <hip/hip_runtime.h>
#include <math.h>

// ---------------------------------------------------------------------------
// STFT as fused GEMM + mag/angle on CDNA5 (gfx1250), wave32 + V_WMMA_F32_16X16X4_F32
//
// spec[b,f,t] = sum_k basis[f,k] * xpad[b, t*STRIDE + k]   (pad = NFFT each side)
// mag = sqrt(re^2+im^2), angle = atan2(im, re); imag filter row = real row + 513
//
// Compute-bound (~69 GFLOP f32 vs ~13us HBM at 23.3 TB/s; x+basis = 38MB fits
// in 192MB L2). f32 WMMA (K=4) makes LDS fragment bandwidth the limiter, so
// each wave computes a 16Mx32N tile (re+im) -> 4 WMMAs per 4 fragment loads
// (1.0 LDS loads/WMMA instead of 1.5).
// ---------------------------------------------------------------------------

typedef float v2f __attribute__((ext_vector_type(2)));
typedef float v8f __attribute__((ext_vector_type(8)));

#define NFFT    1024
#define STRIDE  256
#define CUTOFF  513        // NFFT/2 + 1
#define NFRAMES 1029       // (262144 + 2*1024 - 1024)/256 + 1
#define NSAMP   262144
#define NBATCH  32

#define M_BLK 32           // real filter rows per block (imag rows paired implicitly)
#define N_BLK 128          // frames per block
#define KC    64           // K chunk staged in LDS
#define LDK   (KC + 4)     // padded LDS row stride (floats) -> conflict-free frag reads

__global__ __launch_bounds__(256)
void stft_wmma_kernel(const float* __restrict__ x,
                      const float* __restrict__ basis,
                      float* __restrict__ out)
{
    __shared__ float lds_are[M_BLK * LDK];   // real basis rows,  32x64 (+pad)  ~8.5KB
    __shared__ float lds_aim[M_BLK * LDK];   // imag basis rows,  32x64 (+pad)  ~8.5KB
    __shared__ float lds_b [N_BLK * LDK];    // x window slabs,  128x64 (+pad) ~34.8KB

    const int tid  = threadIdx.x;
    const int lane = tid & 31;               // wave32
    const int wave = tid >> 5;               // 0..7
    const int wm   = wave >> 2;              // 0..1 : M sub-tile (16 rows each)
    const int wn   = wave & 3;               // 0..3 : N sub-tile (32 frames each)

    const int n0 = blockIdx.x * N_BLK;       // first frame of block
    const int f0 = blockIdx.y * M_BLK;       // first (real) filter row of block
    const int b  = blockIdx.z;               // batch

    const float* __restrict__ xb = x + (size_t)b * NSAMP;

    v8f acc_re0 = {0.f, 0.f, 0.f, 0.f, 0.f, 0.f, 0.f, 0.f};
    v8f acc_im0 = acc_re0;
    v8f acc_re1 = acc_re0;
    v8f acc_im1 = acc_re0;

    // A-matrix 16x4 f32 fragment layout: lanes 0-15 hold K={0,1}, lanes 16-31 K={2,3}
    const int row  = lane & 15;
    const int koff = (lane >> 4) * 2;

    for (int k0 = 0; k0 < NFFT; k0 += KC) {
        // ---- stage A (real + imag basis rows), zero-fill rows >= CUTOFF ----
        #pragma unroll
        for (int q = 0; q < 2; ++q) {
            const int v  = tid + q * 256;    // 512 float4 slots (32 rows x 16)
            const int r  = v >> 4;
            const int t4 = (v & 15) * 4;
            const int f  = f0 + r;
            float4 vre = make_float4(0.f, 0.f, 0.f, 0.f);
            float4 vim = vre;
            if (f < CUTOFF) {
                vre = *(const float4*)(basis + (size_t)f * NFFT + k0 + t4);
                vim = *(const float4*)(basis + (size_t)(f + CUTOFF) * NFFT + k0 + t4);
            }
            *(float4*)&lds_are[r * LDK + t4] = vre;
            *(float4*)&lds_aim[r * LDK + t4] = vim;
        }
        // ---- stage B: strided window slabs of x (implicit im2col, pad = NFFT) ----
        #pragma unroll
        for (int q = 0; q < 8; ++q) {
            const int v  = tid + q * 256;    // 2048 float4 slots (128 frames x 16)
            const int j  = v >> 4;
            const int t4 = (v & 15) * 4;
            const long g = (long)(n0 + j) * STRIDE + k0 + t4 - NFFT;
            float4 val;
            if (g >= 0 && g + 4 <= NSAMP) {
                val = *(const float4*)(xb + g);           // aligned: g % 4 == 0
            } else {
                float e0 = 0.f, e1 = 0.f, e2 = 0.f, e3 = 0.f;
                if (g + 0 >= 0 && g + 0 < NSAMP) e0 = xb[g + 0];
                if (g + 1 >= 0 && g + 1 < NSAMP) e1 = xb[g + 1];
                if (g + 2 >= 0 && g + 2 < NSAMP) e2 = xb[g + 2];
                if (g + 3 >= 0 && g + 3 < NSAMP) e3 = xb[g + 3];
                val = make_float4(e0, e1, e2, e3);
            }
            *(float4*)&lds_b[j * LDK + t4] = val;
        }
        // ---- prefetch next chunk's x slab into cache (global_prefetch_b8) ----
        {
            const int nk = k0 + KC;
            if (nk < NFFT) {
                const int j  = (tid >> 4) * 8;           // sample across 128 frames
                const int t4 = (tid & 15) * 4;
                const long g = (long)(n0 + j) * STRIDE + nk + t4 - NFFT;
                if (g >= 0 && g + 4 <= NSAMP) __builtin_prefetch(xb + g, 0, 0);
            }
        }
        __syncthreads();

        // ---- wave tile 16Mx32N (re+im): 4 WMMAs per 4 fragment loads ----
        const float* __restrict__ ar  = &lds_are[(wm * 16 + row) * LDK];
        const float* __restrict__ ai  = &lds_aim[(wm * 16 + row) * LDK];
        const float* __restrict__ bp0 = &lds_b [(wn * 32      + row) * LDK];
        const float* __restrict__ bp1 = &lds_b [(wn * 32 + 16 + row) * LDK];
        #pragma unroll
        for (int ks = 0; ks < KC / 4; ++ks) {
            const int kk = ks * 4 + koff;
            const v2f a1 = *(const v2f*)(ar  + kk);
            const v2f a2 = *(const v2f*)(ai  + kk);
            const v2f b0 = *(const v2f*)(bp0 + kk);
            const v2f b1 = *(const v2f*)(bp1 + kk);
            acc_re0 = __builtin_amdgcn_wmma_f32_16x16x4_f32(
                          false, a1, false, b0, (short)0, acc_re0, false, false);
            acc_im0 = __builtin_amdgcn_wmma_f32_16x16x4_f32(
                          false, a2, false, b0, (short)0, acc_im0, false, false);
            acc_re1 = __builtin_amdgcn_wmma_f32_16x16x4_f32(
                          false, a1, false, b1, (short)0, acc_re1, false, false);
            acc_im1 = __builtin_amdgcn_wmma_f32_16x16x4_f32(
                          false, a2, false, b1, (short)0, acc_im1, false, false);
        }
        __syncthreads();
    }

    // ---- fused epilogue: mag/angle from paired (re, im) accumulators ----
    // C/D layout: lanes 0-15 -> N=lane, M=vgpr;  lanes 16-31 -> N=lane-16, M=vgpr+8
    const int fbase = f0 + wm * 16 + (lane >> 4) * 8;
    float* __restrict__ omag = out;
    float* __restrict__ oang = out + (size_t)NBATCH * CUTOFF * NFRAMES;
    #pragma unroll
    for (int nt = 0; nt < 2; ++nt) {
        const int ncol = n0 + wn * 32 + nt * 16 + (lane & 15);
        if (ncol < NFRAMES) {
            #pragma unroll
            for (int r = 0; r < 8; ++r) {
                const int f = fbase + r;
                if (f < CUTOFF) {
                    const float re  = nt ? acc_re1[r] : acc_re0[r];
                    const float im  = nt ? acc_im1[r] : acc_im0[r];
                    const float mag = sqrtf(re * re + im * im);
                    const float ang = atan2f(im, re);
                    const size_t o  = ((size_t)b * CUTOFF + f) * (size_t)NFRAMES + ncol;
                    omag[o] = mag;
                    oang[o] = ang;
                }
            }
        }
    }
}

extern "C" void kernel_launch(void* const* d_in, const int* in_sizes, int n_in,
                              void* d_out, int out_size, void* d_ws, size_t ws_size,
                              hipStream_t stream)
{
    const float* x     = (const float*)d_in[0];   // (32, 1, 262144) f32
    const float* basis = (const float*)d_in[1];   // (1026, 1, 1024) f32
    float*       out   = (float*)d_out;           // mag || angle, each (32,513,1029) f32

    dim3 grid((NFRAMES + N_BLK - 1) / N_BLK,      // 9 frame blocks
              (CUTOFF  + M_BLK - 1) / M_BLK,      // 17 filter blocks
              NBATCH);                            // 32 batches
    stft_wmma_kernel<<<grid, 256, 0, stream>>>(x, basis, out);
}